// DynamicFrequencyAttention_58875411694360
// MI455X (gfx1250) — compile-verified
//
#include <hip/hip_runtime.h>
#include <hip/hip_bf16.h>
#include <math.h>

// ---------------------------------------------------------------------------
// DynamicFrequencyAttention on MI455X (gfx1250).
// Math: out = sigmoid(MLP(mean_HW(x))) [b,c] * x[b,c,h,w]  (FFT cancels:
//   |F| * e^{i angle(F)} == F, and ifft2(mask*F) == mask * x for scalar mask).
// Memory-bound: ~1.2GB HBM traffic -> ~52us at 23.3 TB/s; compute negligible.
// MLP runs in a single wave32 using v_wmma_f32_16x16x4_f32 (full f32 GEMM).
// ---------------------------------------------------------------------------

#define B_    8
#define C_    192
#define HW_   65536              // 256*256
#define PLANES_ (B_*C_)          // 1536
#define NVEC_ (B_*C_*HW_/4)      // 25,165,824 float4
#define XFREQ_ELEMS_ ((size_t)B_*C_*HW_)  // 100,663,296

typedef __attribute__((ext_vector_type(2))) float v2f;
typedef __attribute__((ext_vector_type(4))) float v4f;
typedef __attribute__((ext_vector_type(8))) float v8f;

// -------------------- Kernel 1: channel mean pooling -----------------------
__global__ __launch_bounds__(256) void pool_kernel(const v4f* __restrict__ x4,
                                                   float* __restrict__ pooled) {
    const int plane = blockIdx.x;                       // b*C + c
    const v4f* p = x4 + (size_t)plane * (HW_ / 4);
    float s = 0.f;
    for (int i = threadIdx.x; i < HW_ / 4; i += 256) {
        v4f v = p[i];
        s += (v.x + v.y) + (v.z + v.w);
    }
    // wave32 reduce
    #pragma unroll
    for (int off = 16; off > 0; off >>= 1) s += __shfl_down(s, off, 32);
    __shared__ float wsum[8];
    if ((threadIdx.x & 31) == 0) wsum[threadIdx.x >> 5] = s;
    __syncthreads();
    if (threadIdx.x == 0) {
        float t = 0.f;
        #pragma unroll
        for (int w = 0; w < 8; ++w) t += wsum[w];
        pooled[plane] = t * (1.0f / (float)HW_);
    }
}

// -------------------- Kernel 2: single-wave WMMA MLP -----------------------
__device__ __forceinline__ float gelu_exact(float x) {
    return 0.5f * x * (1.0f + erff(x * 0.70710678118654752440f));
}
__device__ __forceinline__ float sigmoid_(float x) {
    return 1.0f / (1.0f + expf(-x));
}

// One 16xN GEMM stage on a single wave using V_WMMA_F32_16X16X4_F32.
//   A: LDS, row-major [16][Ka]  (Ka multiple of 4, zero-padded)
//   B: global, row-major [Kb][Nb]  (rows >= Kb treated as 0)
//   Out: row stride No, columns [0,Nb) rows [0,RowsValid) written
// All dims are compile-time so OOB guards fold away where statically true.
// ACT: 0=none 1=gelu 2=sigmoid
template <int ACT, int Ka, int Nb, int Kb, int No, int RowsValid>
__device__ void mlp_stage(const float* __restrict__ A,
                          const float* __restrict__ B,
                          const float* __restrict__ bias,
                          float* __restrict__ Out) {
    const int lane = threadIdx.x;
    const int lo = lane & 15;
    const int hi = lane >> 4;
    #pragma unroll 1
    for (int n0 = 0; n0 < Nb; n0 += 16) {
        const int col = n0 + lo;
        const bool colok = (Nb % 16 == 0) || (col < Nb);
        v8f acc = {};
        #pragma unroll 4
        for (int k0 = 0; k0 < Ka; k0 += 4) {
            const int ka = k0 + 2 * hi;
            v2f a, b;
            a.x = A[lo * Ka + ka];
            a.y = A[lo * Ka + ka + 1];
            b.x = (colok && (Kb >= Ka || ka     < Kb)) ? B[ka * Nb + col]       : 0.f;
            b.y = (colok && (Kb >= Ka || ka + 1 < Kb)) ? B[(ka + 1) * Nb + col] : 0.f;
            acc = __builtin_amdgcn_wmma_f32_16x16x4_f32(
                /*neg_a=*/false, a, /*neg_b=*/false, b,
                /*c_mod=*/(short)0, acc, /*reuse_a=*/false, /*reuse_b=*/false);
        }
        const float bv = colok ? bias[col] : 0.f;
        #pragma unroll
        for (int v = 0; v < 8; ++v) {
            float d = acc[v] + bv;
            if (ACT == 1) d = gelu_exact(d);
            else if (ACT == 2) d = sigmoid_(d);
            const int m = v + 8 * hi;
            if (colok && (RowsValid == 16 || m < RowsValid)) Out[m * No + col] = d;
        }
    }
}

__global__ __launch_bounds__(32) void mlp_kernel(
    const float* __restrict__ pooled,
    const float* __restrict__ W1, const float* __restrict__ b1,
    const float* __restrict__ W2, const float* __restrict__ b2,
    const float* __restrict__ M1, const float* __restrict__ bm1,
    const float* __restrict__ M2, const float* __restrict__ bm2,
    float* __restrict__ mask_out,      // [8*192]
    float* __restrict__ degprob_out) { // [8*3]
    __shared__ float Apad[16 * 192];   // stage-1 A (pooled, zero-padded rows)
    __shared__ float Hbuf[16 * 128];   // stage-1 out
    __shared__ float Dbuf[16 * 4];     // stage-2 out (deg_prob, K-padded)
    __shared__ float Mbuf[16 * 64];    // stage-3 out
    const int tid = threadIdx.x;

    for (int i = tid; i < 16 * 192; i += 32)
        Apad[i] = (i < B_ * C_) ? pooled[i] : 0.f;
    for (int i = tid; i < 16 * 4; i += 32) Dbuf[i] = 0.f;
    __syncthreads();

    // h = gelu(pooled @ W1 + b1)            (16x192)*(192x128)
    mlp_stage<1, 192, 128, 192, 128, 16>(Apad, W1, b1, Hbuf);
    __syncthreads();
    // deg_prob = sigmoid(h @ W2 + b2)       (16x128)*(128x3), out K-padded col3=0
    mlp_stage<2, 128, 3, 128, 4, 16>(Hbuf, W2, b2, Dbuf);
    __syncthreads();
    // emit deg_prob rows 0..7
    for (int i = tid; i < B_ * 3; i += 32)
        degprob_out[i] = Dbuf[(i / 3) * 4 + (i % 3)];
    // m = gelu(deg_prob @ M1 + bm1)         (16x4)*(3x64), B row 3 -> 0
    mlp_stage<1, 4, 64, 3, 64, 16>(Dbuf, M1, bm1, Mbuf);
    __syncthreads();
    // mask = sigmoid(m @ M2 + bm2)          (16x64)*(64x192) -> rows 0..7
    mlp_stage<2, 64, 192, 64, 192, B_>(Mbuf, M2, bm2, mask_out);
}

// -------------------- Kernel 3: out = mask[b,c] * x ------------------------
// Pure stream: x will not be reused, out is never re-read -> non-temporal
// on both sides so we don't thrash the 192MB L2.
__global__ __launch_bounds__(256) void scale_kernel(const v4f* __restrict__ x4,
                                                    const float* __restrict__ mask,
                                                    v4f* __restrict__ out4) {
    const int stride = gridDim.x * blockDim.x;
    for (int i = blockIdx.x * blockDim.x + threadIdx.x; i < NVEC_; i += stride) {
        const float m = mask[i >> 14];  // 16384 float4 per (b,c) plane
        v4f v = __builtin_nontemporal_load(&x4[i]);
        v *= m;
        __builtin_nontemporal_store(v, &out4[i]);
    }
}

// ---------------------------------------------------------------------------
extern "C" void kernel_launch(void* const* d_in, const int* in_sizes, int n_in,
                              void* d_out, int out_size, void* d_ws, size_t ws_size,
                              hipStream_t stream) {
    (void)in_sizes; (void)n_in; (void)out_size; (void)ws_size;
    const float* x   = (const float*)d_in[0];
    const float* W1  = (const float*)d_in[1];
    const float* b1  = (const float*)d_in[2];
    const float* W2  = (const float*)d_in[3];
    const float* b2  = (const float*)d_in[4];
    const float* M1  = (const float*)d_in[5];
    const float* bm1 = (const float*)d_in[6];
    const float* M2  = (const float*)d_in[7];
    const float* bm2 = (const float*)d_in[8];

    float* out         = (float*)d_out;
    float* degprob_out = out + XFREQ_ELEMS_;   // tuple: (x_freq, deg_prob)
    float* pooled      = (float*)d_ws;         // 1536 floats
    float* mask        = pooled + PLANES_;     // 1536 floats

    pool_kernel<<<PLANES_, 256, 0, stream>>>((const v4f*)x, pooled);
    mlp_kernel<<<1, 32, 0, stream>>>(pooled, W1, b1, W2, b2, M1, bm1, M2, bm2,
                                     mask, degprob_out);
    scale_kernel<<<8192, 256, 0, stream>>>((const v4f*)x, mask, (v4f*)out);
}